// VoronoiPropagation_65266323030069
// MI455X (gfx1250) — compile-verified
//
#include <hip/hip_runtime.h>
#include <math.h>

// ---------------------------------------------------------------------------
// VoronoiPropagation for MI455X (gfx1250, wave32).
// Memory-bound pipeline (~1.2 GB traffic => ~50us at 23.3 TB/s HBM).
// fdist uses full-FP32 WMMA (V_WMMA_F32_16X16X4_F32): Gram-diagonal gives
// sum(f^2), centroid-broadcast B gives dot(f,m); fdist = sqrt(S2-2*dot+|m|^2).
// ---------------------------------------------------------------------------

#define BATCH  32
#define KCLUS  56
#define HH     224
#define WWID   224
#define HWPIX  (HH * WWID)        // 50176
#define FEATC  64
#define CP1    65
#define POOLN  (KCLUS * KCLUS)    // 3136

typedef __attribute__((ext_vector_type(2))) float v2f;
typedef __attribute__((ext_vector_type(8))) float v8f;

// ---------------------------------------------------------------------------
// Kernel 1: Sobel(zero-pad 3x3 cross-correlation) -> inverted -> 4x4 mean pool
// One thread per pooled cell (B*56*56 cells); edge map is only 6.4 MB (L2-hot).
// ---------------------------------------------------------------------------
__global__ void k_sobel_pool(const float* __restrict__ in, float* __restrict__ pooled) {
  int t = blockIdx.x * blockDim.x + threadIdx.x;   // 0 .. B*POOLN-1 (exact grid)
  int b = t / POOLN;
  int r = t - b * POOLN;
  int i = r / KCLUS, j = r - i * KCLUS;
  const float* E = in + ((size_t)b * CP1 + FEATC) * HWPIX;  // channel 64 = edges
  int h0 = i * 4, w0 = j * 4;
  float sum = 0.f;
  for (int by = 0; by < 4; ++by) {
    int h = h0 + by;
    for (int bx = 0; bx < 4; ++bx) {
      int w = w0 + bx;
      float p[3][3];
#pragma unroll
      for (int dh = 0; dh < 3; ++dh)
#pragma unroll
        for (int dw = 0; dw < 3; ++dw) {
          int hh = h + dh - 1, ww = w + dw - 1;
          p[dh][dw] = (hh >= 0 && hh < HH && ww >= 0 && ww < WWID)
                          ? E[hh * WWID + ww] : 0.f;
        }
      float gx = (p[0][2] + 2.f * p[1][2] + p[2][2]) - (p[0][0] + 2.f * p[1][0] + p[2][0]);
      float gy = (p[2][0] + 2.f * p[2][1] + p[2][2]) - (p[0][0] + 2.f * p[0][1] + p[0][2]);
      sum += 1.f - 0.5f * sqrtf(gx * gx + gy * gy);
    }
  }
  pooled[t] = sum * (1.f / 16.f);
}

// ---------------------------------------------------------------------------
// Kernel 2: centroid[b][c] = mean(features[b,c,:,:]) - 1e-6  (m-vector)
// One block per (b,c); coalesced grid-stride read of a contiguous 200KB plane.
// ---------------------------------------------------------------------------
__global__ void k_centroid(const float* __restrict__ in, float* __restrict__ m) {
  __shared__ float red[256];
  int b = blockIdx.x >> 6;        // / 64
  int c = blockIdx.x & 63;
  const float* F = in + ((size_t)b * CP1 + c) * HWPIX;
  float s = 0.f;
  for (int i = threadIdx.x; i < HWPIX; i += 256) s += F[i];
  red[threadIdx.x] = s;
  __syncthreads();
  for (int st = 128; st > 0; st >>= 1) {
    if (threadIdx.x < st) red[threadIdx.x] += red[threadIdx.x + st];
    __syncthreads();
  }
  if (threadIdx.x == 0) m[b * FEATC + c] = red[0] * (1.f / (float)HWPIX) - 1e-6f;
}

// ---------------------------------------------------------------------------
// Kernel 3: per-batch top-56 of 3136 via 56 iterative argmax passes in LDS.
// Tie-break: lowest index wins (matches jax.lax.top_k). Emits sy/sx directly.
// ---------------------------------------------------------------------------
__global__ void k_topk(const float* __restrict__ pooled,
                       float* __restrict__ sy, float* __restrict__ sx) {
  __shared__ float vals[POOLN];
  __shared__ float rv[256];
  __shared__ int   ri[256];
  int b = blockIdx.x, tid = threadIdx.x;
  for (int i = tid; i < POOLN; i += 256) vals[i] = pooled[b * POOLN + i];
  __syncthreads();
  for (int k = 0; k < KCLUS; ++k) {
    float best = -3.402823466e38f; int bi = POOLN;
    for (int i = tid; i < POOLN; i += 256) {          // increasing i: strict >
      float v = vals[i];                               // keeps lowest index
      if (v > best) { best = v; bi = i; }
    }
    rv[tid] = best; ri[tid] = bi;
    __syncthreads();
    for (int st = 128; st > 0; st >>= 1) {
      if (tid < st) {
        float v2 = rv[tid + st]; int i2 = ri[tid + st];
        if (v2 > rv[tid] || (v2 == rv[tid] && i2 < ri[tid])) { rv[tid] = v2; ri[tid] = i2; }
      }
      __syncthreads();
    }
    if (tid == 0) {
      int idx = ri[0];
      sy[b * KCLUS + k] = (float)(idx / KCLUS) * (1.f / KCLUS);
      sx[b * KCLUS + k] = (float)(idx % KCLUS) * (1.f / KCLUS);
      vals[idx] = -3.402823466e38f;
    }
    __syncthreads();
  }
}

// ---------------------------------------------------------------------------
// Kernel 4: fdist via FP32 WMMA. Each wave owns one 16-pixel tile.
//   A[m,k] = f[c_base+k, p_m]  (16x4 fp32: v0 = K0|K2 per lane-half, v1 = K1|K3)
//   B-gram = same registers as A (B 4x16 layout: row0|row2 in v0, row1|row3 in v1)
//   B-dot  = centroid broadcast across N.
//   D_gram[p][p] = sum_c f^2 (diagonal), D_dot[p][*] = dot(f_p, m).
// 16 chained WMMAs per accumulator cover K=64. Extraction via per-wave LDS.
// ---------------------------------------------------------------------------
__global__ void k_fdist_wmma(const float* __restrict__ in, const float* __restrict__ m,
                             float* __restrict__ fd) {
  __shared__ float m_s[FEATC];
  __shared__ float msq_s;
  __shared__ float sg[8][8][32];
  __shared__ float sd[8][8][32];
  int tid  = threadIdx.x;
  int wave = tid >> 5, lane = tid & 31;
  int tile  = blockIdx.x * 8 + wave;      // 3136 tiles per batch; 392 blocks/batch
  int b     = blockIdx.x / 392;           // all 8 waves in a block share b
  int pbase = (tile - b * 3136) * 16;

  if (tid < FEATC) m_s[tid] = m[b * FEATC + tid];
  __syncthreads();
  if (tid == 0) {
    float s = 0.f;
    for (int c = 0; c < FEATC; ++c) s += m_s[c] * m_s[c];
    msq_s = s;
  }
  __syncthreads();

  const float* Fb = in + (size_t)b * CP1 * HWPIX;
  int px    = pbase + (lane & 15);        // pixel index within batch plane
  int chalf = (lane >> 4) << 1;           // 0 for lanes 0-15, 2 for lanes 16-31
  __builtin_prefetch(Fb + (size_t)chalf * HWPIX + px, 0, 0);  // global_prefetch_b8

  v8f gram = {};
  v8f dacc = {};
#pragma unroll
  for (int ch = 0; ch < 16; ++ch) {
    int c0 = ch * 4 + chalf;
    v2f a, bm;
    a[0]  = Fb[(size_t)c0 * HWPIX + px];          // coalesced 64B per lane-half
    a[1]  = Fb[(size_t)(c0 + 1) * HWPIX + px];
    bm[0] = m_s[c0];
    bm[1] = m_s[c0 + 1];
    // (neg_a, A, neg_b, B, c_mod, C, reuse_a, reuse_b)
    gram = __builtin_amdgcn_wmma_f32_16x16x4_f32(false, a, false, a,  (short)0, gram, false, false);
    dacc = __builtin_amdgcn_wmma_f32_16x16x4_f32(false, a, false, bm, (short)0, dacc, false, false);
  }

#pragma unroll
  for (int r2 = 0; r2 < 8; ++r2) {
    sg[wave][r2][lane] = gram[r2];
    sd[wave][r2][lane] = dacc[r2];
  }
  __syncthreads();

  if (lane < 16) {
    int p  = lane;
    int r2 = p & 7;
    int li = (p < 8) ? p : (p + 16);   // C/D layout: M=r|M=8+r split on lane halves
    float S2 = sg[wave][r2][li];
    float dv = sd[wave][r2][li];
    float d2 = S2 - 2.f * dv + msq_s;
    fd[(size_t)b * HWPIX + pbase + p] = sqrtf(fmaxf(d2, 0.f));
  }
}

// ---------------------------------------------------------------------------
// Kernel 5: combined = spatial + fdist; softmax over K=56; 360 MB of coalesced
// strided stores. comb[56] lives in registers (fully unrolled, ~80 VGPRs).
// ---------------------------------------------------------------------------
__global__ void k_markers(const float* __restrict__ fd, const float* __restrict__ sy,
                          const float* __restrict__ sx, const float* __restrict__ stdp,
                          float* __restrict__ out) {
  __shared__ float sy_s[KCLUS], sx_s[KCLUS];
  int tid = threadIdx.x;
  int b   = blockIdx.x / 196;                 // 196 blocks per batch (exact)
  int p   = (blockIdx.x - b * 196) * 256 + tid;
  if (tid < KCLUS) { sy_s[tid] = sy[b * KCLUS + tid]; sx_s[tid] = sx[b * KCLUS + tid]; }
  __syncthreads();

  float inv_std = 1.f / stdp[0];
  float fdv = fd[(size_t)b * HWPIX + p];
  int h = p / WWID, w = p - h * WWID;
  float y = (float)h * (1.f / HH);
  float x = (float)w * (1.f / WWID);

  float comb[KCLUS];
  float mx = -3.402823466e38f;
#pragma unroll
  for (int k = 0; k < KCLUS; ++k) {
    float dy = (y - sy_s[k]) * inv_std;
    float dx = (x - sx_s[k]) * inv_std;
    float s  = 0.5f * (__expf(-dy * dy) + __expf(-dx * dx));
    float cv = s + fdv;
    comb[k] = cv;
    mx = fmaxf(mx, cv);
  }
  float sum = 0.f;
#pragma unroll
  for (int k = 0; k < KCLUS; ++k) {
    float e = __expf(comb[k] - mx);
    comb[k] = e;
    sum += e;
  }
  float inv = 1.f / sum;
  size_t ob = (size_t)b * KCLUS * HWPIX + p;
#pragma unroll
  for (int k = 0; k < KCLUS; ++k)
    out[ob + (size_t)k * HWPIX] = comb[k] * inv;   // coalesced per-k stores
}

// ---------------------------------------------------------------------------
// Launch: 5 kernels on one stream (stream order = dependency order).
// Workspace: pooled(100352) + m(2048) + sy(1792) + sx(1792) + fd(1605632)
// floats = ~6.9 MB, all written before read (ws is poisoned by harness).
// ---------------------------------------------------------------------------
extern "C" void kernel_launch(void* const* d_in, const int* in_sizes, int n_in,
                              void* d_out, int out_size, void* d_ws, size_t ws_size,
                              hipStream_t stream) {
  const float* in   = (const float*)d_in[0];   // (32, 65, 224, 224) fp32
  const float* stdp = (const float*)d_in[1];   // scalar
  float* out = (float*)d_out;                  // (32, 56, 224, 224) fp32
  float* ws  = (float*)d_ws;

  float* pooled = ws;                            // B*POOLN
  float* m      = pooled + BATCH * POOLN;        // B*64
  float* sy     = m + BATCH * FEATC;             // B*56
  float* sx     = sy + BATCH * KCLUS;            // B*56
  float* fd     = sx + BATCH * KCLUS;            // B*HWPIX

  k_sobel_pool<<<(BATCH * POOLN) / 256, 256, 0, stream>>>(in, pooled);
  k_centroid  <<<BATCH * FEATC,         256, 0, stream>>>(in, m);
  k_topk      <<<BATCH,                 256, 0, stream>>>(pooled, sy, sx);
  k_fdist_wmma<<<(BATCH * POOLN) / 8,   256, 0, stream>>>(in, m, fd);
  k_markers   <<<(BATCH * HWPIX) / 256, 256, 0, stream>>>(fd, sy, sx, stdp, out);
}